// Net_3083786518568
// MI455X (gfx1250) — compile-verified
//
#include <hip/hip_runtime.h>
#include <hip/hip_bf16.h>

// CDNA5 / gfx1250: wave32, WMMA f32 16x16x4 for the f32 RNN recurrence.
typedef float v2f __attribute__((ext_vector_type(2)));
typedef float v8f __attribute__((ext_vector_type(8)));

#define RNN_H 10   // hidden size == vocab == out features

// ds_swizzle group-of-32 pattern SWAPX16: xor_mask=0x10, or=0, and=0x1f
// -> lanes exchange values with lane^16; pattern is an instruction immediate,
// so no index VGPR and no VALU setup (unlike __shfl_xor's clamp/cndmask).
__device__ __forceinline__ float swap16(float v) {
  return __int_as_float(__builtin_amdgcn_ds_swizzle(__float_as_int(v), 0x401F));
}

// One wave (32 lanes) owns a tile of 16 batch rows across all S timesteps.
//
// Orientation:  D[h][b] = Whh(M=h, K=k) x Hprev^T(K=k, N=b) + Xp[h][b]
//   - A operands (w_hh / fc_w) are loop-invariant register fragments.
//   - Recurrent state is carried as the B operand (K=hidden, N=batch).
//   - D-layout -> next B-layout needs only 6 ds_swizzle SWAPX16 moves.
// K=10 padded to 12 -> 3 chained V_WMMA_F32_16X16X4_F32 per matmul.
//
// 2-deep software pipeline: token(t+2) is loaded during step t; e(t+1) and
// Xp(t+1) are computed during step t (overlapped with the relayout swizzles
// and FC WMMAs). Critical path per step:
//   3x v_wmma -> v_max_num (in-place asm ReLU) -> ds_swizzle -> cndmask -> wmma
__global__ __launch_bounds__(32) void rnn_wmma_kernel(
    const int* __restrict__ x, const float* __restrict__ h0,
    const float* __restrict__ em, const float* __restrict__ w_ih,
    const float* __restrict__ w_hh, const float* __restrict__ b_ih,
    const float* __restrict__ b_hh, const float* __restrict__ fc_w,
    const float* __restrict__ fc_b, float* __restrict__ y,
    float* __restrict__ hlast, int B, int S)
{
  const int lane = threadIdx.x & 31;
  const int half = lane >> 4;     // 0: K=4c,4c+1   1: K=4c+2,4c+3 (A/B frag halves)
  const int lm   = lane & 15;     // A: row M; B/C/D: column N
  const int base = blockIdx.x * 16;
  if (base >= B) return;

  // ---- loop-invariant A fragments: A[m][k], lane m holds K pair per chunk ----
  v2f aW[3], aF[3];
#pragma unroll
  for (int c = 0; c < 3; ++c) {
    const int k0 = 4 * c + 2 * half;
    float w0 = 0.f, w1 = 0.f, f0 = 0.f, f1 = 0.f;
    if (lm < RNN_H) {
      if (k0     < RNN_H) { w0 = w_hh[lm * RNN_H + k0];     f0 = fc_w[lm * RNN_H + k0]; }
      if (k0 + 1 < RNN_H) { w1 = w_hh[lm * RNN_H + k0 + 1]; f1 = fc_w[lm * RNN_H + k0 + 1]; }
    }
    aW[c].x = w0; aW[c].y = w1;
    aF[c].x = f0; aF[c].y = f1;
  }

  // ---- per-lane C/D-layout constants: vgpr v <-> row M = v + 8*half ----
  float wihv[8], biasv[8], fcbv[8];
#pragma unroll
  for (int v = 0; v < 8; ++v) {
    const int hh = v + 8 * half;
    const bool ok = hh < RNN_H;
    wihv[v]  = ok ? w_ih[hh] : 0.f;
    biasv[v] = ok ? (b_ih[hh] + b_hh[hh]) : 0.f;   // b_ih + b_hh fused
    fcbv[v]  = ok ? fc_b[hh] : 0.f;
  }

  // Embedding table distributed across lanes 0..9; gathered via ds_bpermute.
  const int emreg = __float_as_int((lane < RNN_H) ? em[lane] : 0.f);

  // ---- initial state as B fragments: B[k][b] = h0[b][k] ----
  v2f hb[3];
#pragma unroll
  for (int c = 0; c < 3; ++c) {
    const int k0 = 4 * c + 2 * half;
    hb[c].x = (k0     < RNN_H) ? h0[(size_t)(base + lm) * RNN_H + k0]     : 0.f;
    hb[c].y = (k0 + 1 < RNN_H) ? h0[(size_t)(base + lm) * RNN_H + k0 + 1] : 0.f;
  }

  const size_t xrow = (size_t)(base + lm) * S;   // this lane's token stream

  // ---- 2-deep pipeline prologue ----
  // cxp holds Xp for the step about to execute; tokIdx1 is the byte-index
  // (tok<<2; bpermute only reads bits [6:2]) for the step after that.
  const int tok0 = x[xrow] << 2;
  int tokIdx1 = x[xrow + ((1 < S) ? 1 : 0)] << 2;
  v8f cxp;
  {
    const float e0 = __int_as_float(__builtin_amdgcn_ds_bpermute(tok0, emreg));
#pragma unroll
    for (int v = 0; v < 8; ++v) cxp[v] = fmaf(e0, wihv[v], biasv[v]);
  }

  v8f d = {};                                    // D-layout hidden state (post-ReLU)
  for (int t = 0; t < S; ++t) {
    // token fetch for t+2 (two full iterations of latency slack)
    const int tn = (t + 2 < S) ? (t + 2) : (S - 1);
    const int tokIdx2 = x[xrow + tn] << 2;
    if ((t & 63) == 0) {
      const int tp = (t + 128 < S) ? (t + 128) : (S - 1);
      __builtin_prefetch(&x[xrow + tp], 0, 1);   // global_prefetch_b8
    }

    // h_new = relu(Whh x Hprev^T + Xp): 3 chained f32 WMMAs (K = 0..11)
    d = __builtin_amdgcn_wmma_f32_16x16x4_f32(false, aW[0], false, hb[0], (short)0, cxp, false, false);
    d = __builtin_amdgcn_wmma_f32_16x16x4_f32(false, aW[1], false, hb[1], (short)0, d,   false, false);
    d = __builtin_amdgcn_wmma_f32_16x16x4_f32(false, aW[2], false, hb[2], (short)0, d,   false, false);
    // in-place single-instruction ReLU (tied operand: no spill copies,
    // no fmaxf canonicalize prefix)
#pragma unroll
    for (int v = 0; v < 8; ++v)
      asm("v_max_num_f32 %0, 0, %0" : "+v"(d[v]));

    // D-layout (M=h striped over vgprs/halves) -> B-layout (lane=b, vgprs=K pair)
    // lanes<16 need K={0,1},{4,5},{8,9}; lanes>=16 need K={2,3},{6,7},{0,0}
    const float s0 = swap16(d[0]);
    const float s1 = swap16(d[1]);
    const float s2 = swap16(d[2]);
    const float s3 = swap16(d[3]);
    const float s6 = swap16(d[6]);
    const float s7 = swap16(d[7]);
    if (half == 0) {
      hb[0].x = d[0]; hb[0].y = d[1];   // K=0,1
      hb[1].x = d[4]; hb[1].y = d[5];   // K=4,5
      hb[2].x = s0;   hb[2].y = s1;     // K=8,9 (live in upper half's d[0],d[1])
    } else {
      hb[0].x = s2;   hb[0].y = s3;     // K=2,3 (live in lower half's d[2],d[3])
      hb[1].x = s6;   hb[1].y = s7;     // K=6,7
      hb[2].x = 0.f;  hb[2].y = 0.f;    // K=10,11 zero padding
    }

    // y_t = Fcw x Hnew^T + fc_b  — reuses fresh hb fragments, off critical path
    v8f cfc;
#pragma unroll
    for (int v = 0; v < 8; ++v) cfc[v] = fcbv[v];
    v8f yv;
    yv = __builtin_amdgcn_wmma_f32_16x16x4_f32(false, aF[0], false, hb[0], (short)0, cfc, false, false);
    yv = __builtin_amdgcn_wmma_f32_16x16x4_f32(false, aF[1], false, hb[1], (short)0, yv,  false, false);
    yv = __builtin_amdgcn_wmma_f32_16x16x4_f32(false, aF[2], false, hb[2], (short)0, yv,  false, false);

    // prepare Xp for t+1 now (token was loaded during t-1, long since landed);
    // this bpermute shares the s_wait_dscnt the relayout swizzles already pay.
    {
      const float e = __int_as_float(__builtin_amdgcn_ds_bpermute(tokIdx1, emreg));
#pragma unroll
      for (int v = 0; v < 8; ++v) cxp[v] = fmaf(e, wihv[v], biasv[v]);
    }

    // store y[b][t][o]: lane=b; lower half o=0..7, upper half o=8,9 (8B-aligned)
    float* yp = y + ((size_t)(base + lm) * S + t) * RNN_H;
    if (half == 0) {
#pragma unroll
      for (int v = 0; v < 8; v += 2)
        *(float2*)(yp + v) = make_float2(yv[v], yv[v + 1]);
    } else {
      *(float2*)(yp + 8) = make_float2(yv[0], yv[1]);
    }

    tokIdx1 = tokIdx2;
  }

  // h_last[b][h] from final D-layout state
  float* hp = hlast + (size_t)(base + lm) * RNN_H;
  if (half == 0) {
#pragma unroll
    for (int v = 0; v < 8; v += 2)
      *(float2*)(hp + v) = make_float2(d[v], d[v + 1]);
  } else {
    *(float2*)(hp + 8) = make_float2(d[0], d[1]);
  }
}

extern "C" void kernel_launch(void* const* d_in, const int* in_sizes, int n_in,
                              void* d_out, int out_size, void* d_ws, size_t ws_size,
                              hipStream_t stream) {
  const int*   x    = (const int*)  d_in[0];  // [B,S] tokens
  const float* h    = (const float*)d_in[1];  // [1,B,H]
  const float* em   = (const float*)d_in[2];  // [V,1]
  const float* w_ih = (const float*)d_in[3];  // [H,1]
  const float* w_hh = (const float*)d_in[4];  // [H,H]
  const float* b_ih = (const float*)d_in[5];  // [H]
  const float* b_hh = (const float*)d_in[6];  // [H]
  const float* fc_w = (const float*)d_in[7];  // [H,H]
  const float* fc_b = (const float*)d_in[8];  // [H]

  const int B = in_sizes[1] / RNN_H;          // h flat = B*H
  const int S = in_sizes[0] / B;              // x flat = B*S

  float* y     = (float*)d_out;               // [B,S,H]
  float* hlast = y + (size_t)B * S * RNN_H;   // [1,B,H] appended

  const int tiles = (B + 15) / 16;            // 128 single-wave workgroups
  rnn_wmma_kernel<<<dim3(tiles), dim3(32), 0, stream>>>(
      x, h, em, w_ih, w_hh, b_ih, b_hh, fc_w, fc_b, y, hlast, B, S);
}